// BlockSpinSampler_85504208929061
// MI455X (gfx1250) — compile-verified
//
#include <hip/hip_runtime.h>
#include <stdint.h>

// ---- problem constants (from reference) ----
#define Hn 4096   // nodes per block
#define Nn 8192   // total nodes
#define Cn 2048   // chains

typedef __attribute__((ext_vector_type(2))) float v2f;
typedef __attribute__((ext_vector_type(8))) float v8f;

// ---------------- jax-exact threefry2x32 ----------------
__host__ __device__ inline uint32_t rotl32(uint32_t x, uint32_t r) {
  return (x << r) | (x >> (32u - r));
}

__host__ __device__ inline void threefry2x32(uint32_t k0, uint32_t k1,
                                             uint32_t c0, uint32_t c1,
                                             uint32_t& o0, uint32_t& o1) {
  const uint32_t k2 = k0 ^ k1 ^ 0x1BD11BDAu;
  uint32_t x0 = c0 + k0, x1 = c1 + k1;
#define TF_R(r) { x0 += x1; x1 = rotl32(x1, r); x1 ^= x0; }
  TF_R(13) TF_R(15) TF_R(26) TF_R(6)
  x0 += k1; x1 += k2 + 1u;
  TF_R(17) TF_R(29) TF_R(16) TF_R(24)
  x0 += k2; x1 += k0 + 2u;
  TF_R(13) TF_R(15) TF_R(26) TF_R(6)
  x0 += k0; x1 += k1 + 3u;
  TF_R(17) TF_R(29) TF_R(16) TF_R(24)
  x0 += k1; x1 += k2 + 4u;
  TF_R(13) TF_R(15) TF_R(26) TF_R(6)
  x0 += k2; x1 += k0 + 5u;
#undef TF_R
  o0 = x0; o1 = x1;
}

// banded coupling value for the WMMA B-matrix (zero outside the 8-wide band);
// index-clamp (v_med3) + select keeps EXEC all-ones (WMMA requirement).
__device__ inline float band_j(int blk, int j, int node,
                               const float* __restrict__ quad) {
  const int d   = blk ? (node - j) : (j - node);          // k in [0,8) if on band
  const int row = blk ? ((j + Hn) & (Hn - 1)) : node;     // quadratic row
  const int dcl = d < 0 ? 0 : (d > 7 ? 7 : d);
  const float qv = quad[(uint32_t)((row << 3) + dcl)];
  return (d == dcl) ? qv : 0.0f;
}

// One Gibbs half-step: update block `blk` columns of x from the other block.
// One wave per {16 chains, 16 chains+1024} x 16-node field tile pair:
//   - 12 x V_WMMA_F32_16X16X4_F32 (full fp32) over the 24-wide banded window,
//     B (the J band) shared between the two chain tiles
//   - each threefry2x32 yields BOTH jax counter outputs (m and m+2^22),
//     which are exactly the draws for chains c and c+1024 at the same node.
// All state indices are u32 (buffer is 64 MB) so loads/stores lower to the
// cheap GVS form: SGPR base + 32-bit lane offset with scale_offset.
__global__ __launch_bounds__(256)
void spin_halfstep(float* __restrict__ x,
                   const float* __restrict__ quad,
                   const float* __restrict__ lin,
                   const float* __restrict__ sched,
                   int t, int blk, uint32_t sk0, uint32_t sk1) {
  const int lane = threadIdx.x & 31;
  const int wave = threadIdx.x >> 5;
  const int tile = blockIdx.x * 8 + wave;   // 0 .. 16383
  const int tc = tile >> 8;                 // chain-pair tile 0..63
  const int ti = tile & 255;                // node tile 0..255
  const int c0 = tc << 4;                   // chains [c0,c0+16) and +1024
  const int i0 = ti << 4;
  const int l    = lane & 15;
  const int half = lane >> 4;

  const float beta = sched[t];              // scalar (uniform) load
  const uint32_t yoff = blk ? 0u : (uint32_t)Hn;  // read the *other* half
  const uint32_t woff = blk ? (uint32_t)Hn : 0u;  // write this half
  const int      j0   = blk ? (i0 - 7) : i0;      // start of banded j-window

  const int cA = c0 + l;                    // A lane: chain row (M), tile 0
  const int iB = i0 + l;                    // B lane: node column (N)
  const uint32_t rowA0 = (uint32_t)cA * Nn + yoff;            // u32 element idx
  const uint32_t rowA1 = (uint32_t)(cA + 1024) * Nn + yoff;

  __builtin_prefetch(&quad[iB << 3], 0, 1); // global_prefetch_b8

  v8f acc0 = {};                            // chains c0 .. c0+15
  v8f acc1 = {};                            // chains c0+1024 .. c0+1039
  #pragma unroll
  for (int s = 0; s < 6; ++s) {
    const int jbase = j0 + 4 * s + 2 * half;   // this lane's K pair
    const uint32_t jw0 = (uint32_t)((jbase + Hn)     & (Hn - 1));
    const uint32_t jw1 = (uint32_t)((jbase + 1 + Hn) & (Hn - 1));
    v2f a0, a1, bb;
    a0.x = x[rowA0 + jw0];
    a0.y = x[rowA0 + jw1];
    a1.x = x[rowA1 + jw0];
    a1.y = x[rowA1 + jw1];
    bb.x = band_j(blk, jbase,     iB, quad);
    bb.y = band_j(blk, jbase + 1, iB, quad);
    // D(16x16 f32) += A(16x4 f32) x B(4x16 f32); B shared across both tiles
    acc0 = __builtin_amdgcn_wmma_f32_16x16x4_f32(
        false, a0, false, bb, (short)0, acc0, false, false);
    acc1 = __builtin_amdgcn_wmma_f32_16x16x4_f32(
        false, a1, false, bb, (short)0, acc1, false, false);
  }

  // epilogue: bias, sigmoid, jax-exact paired uniform draws, spin writes
  // (lanes 0-15 / 16-31 each cover 16 consecutive columns of one chain row)
  const int   ie = i0 + l;
  const float h  = lin[woff + (uint32_t)ie];
  #pragma unroll
  for (int v = 0; v < 8; ++v) {
    const int c = c0 + v + (half ? 8 : 0);   // C/D layout: M = v (+8 hi lanes)
    const float f0 = acc0[v] + h;            // chain c        (< 1024)
    const float f1 = acc1[v] + h;            // chain c + 1024
    const float p0 = 1.0f / (1.0f + __expf(2.0f * beta * f0));
    const float p1 = 1.0f / (1.0f + __expf(2.0f * beta * f1));

    // jax uniform counter layout for shape (2048, 4096): element m pairs
    // with m + n/2 (n/2 = 2^22); c < 1024 here, so m < 2^22 always.
    const uint32_t m = (uint32_t)c * (uint32_t)Hn + (uint32_t)ie;
    uint32_t b0, b1;
    threefry2x32(sk0, sk1, m, m + (1u << 22), b0, b1);
    const float u0 = __uint_as_float((b0 >> 9) | 0x3F800000u) - 1.0f;
    const float u1 = __uint_as_float((b1 >> 9) | 0x3F800000u) - 1.0f;

    const uint32_t wbase = (uint32_t)c * Nn + woff + (uint32_t)ie;
    x[wbase]                       = (u0 < p0) ? 1.0f : -1.0f;
    x[wbase + 1024u * (uint32_t)Nn] = (u1 < p1) ? 1.0f : -1.0f;
  }
}

extern "C" void kernel_launch(void* const* d_in, const int* in_sizes, int n_in,
                              void* d_out, int out_size, void* d_ws, size_t ws_size,
                              hipStream_t stream) {
  (void)in_sizes; (void)n_in; (void)d_ws; (void)ws_size; (void)out_size;
  const float* x_in  = (const float*)d_in[0];   // (C, N) fp32
  const float* lin   = (const float*)d_in[1];   // (N,)   fp32
  const float* quad  = (const float*)d_in[2];   // (H*8,) fp32
  const float* sched = (const float*)d_in[3];   // (3,)   fp32
  // d_in[4..7] = adj, adj_w, block0, block1: banded structure is analytic,
  // never dereferenced.
  float* out = (float*)d_out;

  // fresh copy of the input state each call (deterministic, input preserved)
  hipMemcpyAsync(out, x_in, (size_t)Cn * Nn * sizeof(float),
                 hipMemcpyDeviceToDevice, stream);

  // 6 globally-ordered half-steps; subkey = fold_in(key(42), 2t+b) is pure
  // integer math -> computed on host, passed as kernargs.
  for (int s = 0; s < 6; ++s) {
    uint32_t sk0, sk1;
    threefry2x32(0u, 42u, 0u, (uint32_t)s, sk0, sk1);
    spin_halfstep<<<2048, 256, 0, stream>>>(out, quad, lin, sched,
                                            s >> 1, s & 1, sk0, sk1);
  }
}